// nn_model_75316546503051
// MI455X (gfx1250) — compile-verified
//
#include <hip/hip_runtime.h>

typedef __attribute__((ext_vector_type(2))) float v2f;
typedef __attribute__((ext_vector_type(8))) float v8f;

#define BM 128
#define BN 64
#define BK 32
#define LDS_STRIDE 36  // padded (floats): conflict-free ds_load_b64 frags, 16B-aligned b128 stores

// ---------------------------------------------------------------------------
// Tiled f32 WMMA GEMM with async-to-LDS double-buffered A staging.
//   C[z] = A[z] @ B[z] + bias[z] (+ addend)
// Block = 256 threads (8 wave32s); block tile 128x64; K-tile 32.
// Waves arranged 4(M) x 2(N); each wave computes a 32x32 C tile with 4
// accumulators -> 4 v_wmma_f32_16x16x4_f32 per K-step, 32 per K-tile.
// bNodeMode=1: block columns map to groups of 4 "nodes": global column
// c -> B + (c>>4)*nodeStrideB + k*ldb + (c&15)   (for the Wg einsum, ldb=16).
// Requires M % 128 == 0, N % 64 == 0, K % 32 == 0 (true for every use here).
// ---------------------------------------------------------------------------
__global__ __launch_bounds__(256) void gemm_wmma_f32_k(
    const float* __restrict__ A, int lda, long strideA,
    const float* __restrict__ B, int ldb, long nodeStrideB, int bNodeMode,
    const float* __restrict__ bias, int strideBias,
    const float* __restrict__ addend, int ldadd,
    float* __restrict__ C, int ldc, long strideC,
    int K)
{
  __shared__ float As[2][BM * LDS_STRIDE];  // 2 x 18 KB
  __shared__ float Bs[2][BN * LDS_STRIDE];  // 2 x 9 KB (transposed: Bs[n][k])

  const int batch = blockIdx.z;
  const float* Ab = A + (long)batch * strideA;
  const float* Bb = bNodeMode ? (B + (long)batch * 4 * nodeStrideB) : B;
  const float* biasb = bias ? (bias + (long)batch * strideBias) : nullptr;
  float* Cb = C + (long)batch * strideC;

  const int row0 = blockIdx.x * BM;
  const int col0 = blockIdx.y * BN;
  const int tid  = threadIdx.x;
  const int wave = tid >> 5;
  const int lane = tid & 31;
  const int lgrp = lane >> 4;   // 0: K+0,K+1   1: K+2,K+3  (ISA 16x4 A / 4x16 B layout)
  const int lrow = lane & 15;   // M index (A frag) / N index (B frag)
  const int wm   = wave & 3;    // wave M position (4 x 32 rows)
  const int wn   = wave >> 2;   // wave N position (2 x 32 cols)

  // A staging coords: 256 threads, float4 each, 4 row-passes of 32
  const int a_k = (tid & 7) << 2;   // 0..28
  const int a_m = tid >> 3;         // 0..31
  // B staging coords: 256 threads, 2 transposed float4 each
  const int b_n  = tid & 63;        // 0..63
  const int b_k0 = (tid >> 6) << 2; // 0,4,8,12

  // Precompute B column base (handles node mode)
  const int bcol = col0 + b_n;
  const float* Bcol = bNodeMode ? (Bb + (long)(bcol >> 4) * nodeStrideB + (bcol & 15))
                                : (Bb + bcol);

  v8f acc[2][2];
  #pragma unroll
  for (int ms = 0; ms < 2; ++ms)
    #pragma unroll
    for (int ns = 0; ns < 2; ++ns)
      acc[ms][ns] = (v8f){0.f, 0.f, 0.f, 0.f, 0.f, 0.f, 0.f, 0.f};

  // ---- prologue: stage K-tile 0 into buffer 0 -----------------------------
  #pragma unroll
  for (int mm = 0; mm < BM; mm += 32) {
    const float* gp = Ab + (long)(row0 + a_m + mm) * lda + a_k;
    const unsigned la = (unsigned)(uintptr_t)(&As[0][(a_m + mm) * LDS_STRIDE + a_k]);
    asm volatile("global_load_async_to_lds_b128 %0, %1, off"
                 :: "v"(la), "v"(gp) : "memory");
  }
  {
    #pragma unroll
    for (int kh = 0; kh < 2; ++kh) {
      const int kb = b_k0 + kh * 16;
      float4 v;
      v.x = Bcol[(long)(kb + 0) * ldb];
      v.y = Bcol[(long)(kb + 1) * ldb];
      v.z = Bcol[(long)(kb + 2) * ldb];
      v.w = Bcol[(long)(kb + 3) * ldb];
      *(float4*)(&Bs[0][b_n * LDS_STRIDE + kb]) = v;
    }
  }
  asm volatile("s_wait_asynccnt 0x0" ::: "memory");
  __syncthreads();

  // ---- main pipelined loop ------------------------------------------------
  int cur = 0;
  for (int k0 = 0; k0 < K; k0 += BK) {
    const int nxt = cur ^ 1;
    const bool has_next = (k0 + BK) < K;

    float4 breg[2];
    if (has_next) {
      const int kn = k0 + BK;
      // kick off async A loads for the next tile (overlap with WMMA below)
      #pragma unroll
      for (int mm = 0; mm < BM; mm += 32) {
        const float* gp = Ab + (long)(row0 + a_m + mm) * lda + (kn + a_k);
        const unsigned la = (unsigned)(uintptr_t)(&As[nxt][(a_m + mm) * LDS_STRIDE + a_k]);
        asm volatile("global_load_async_to_lds_b128 %0, %1, off"
                     :: "v"(la), "v"(gp) : "memory");
      }
      #pragma unroll
      for (int kh = 0; kh < 2; ++kh) {
        const int kb = b_k0 + kh * 16;
        breg[kh].x = Bcol[(long)(kn + kb + 0) * ldb];
        breg[kh].y = Bcol[(long)(kn + kb + 1) * ldb];
        breg[kh].z = Bcol[(long)(kn + kb + 2) * ldb];
        breg[kh].w = Bcol[(long)(kn + kb + 3) * ldb];
      }
    }

    // compute 32 WMMAs from the current buffer
    const float* a0 = &As[cur][(wm * 32 + lrow) * LDS_STRIDE + 2 * lgrp];
    const float* a1 = a0 + 16 * LDS_STRIDE;
    const float* b0 = &Bs[cur][(wn * 32 + lrow) * LDS_STRIDE + 2 * lgrp];
    const float* b1 = b0 + 16 * LDS_STRIDE;
    #pragma unroll
    for (int kk = 0; kk < BK; kk += 4) {
      const v2f af0 = *(const v2f*)(a0 + kk);
      const v2f af1 = *(const v2f*)(a1 + kk);
      const v2f bf0 = *(const v2f*)(b0 + kk);
      const v2f bf1 = *(const v2f*)(b1 + kk);
      acc[0][0] = __builtin_amdgcn_wmma_f32_16x16x4_f32(false, af0, false, bf0, (short)0,
                                                        acc[0][0], false, false);
      acc[0][1] = __builtin_amdgcn_wmma_f32_16x16x4_f32(false, af0, false, bf1, (short)0,
                                                        acc[0][1], false, false);
      acc[1][0] = __builtin_amdgcn_wmma_f32_16x16x4_f32(false, af1, false, bf0, (short)0,
                                                        acc[1][0], false, false);
      acc[1][1] = __builtin_amdgcn_wmma_f32_16x16x4_f32(false, af1, false, bf1, (short)0,
                                                        acc[1][1], false, false);
    }

    if (has_next) {
      #pragma unroll
      for (int kh = 0; kh < 2; ++kh)
        *(float4*)(&Bs[nxt][b_n * LDS_STRIDE + b_k0 + kh * 16]) = breg[kh];
      // each wave waits for its own async writes; barrier then makes ALL
      // waves' writes to the incoming buffer visible block-wide
      asm volatile("s_wait_asynccnt 0x0" ::: "memory");
    }
    __syncthreads();
    cur = nxt;
  }

  // ---- epilogue -----------------------------------------------------------
  const bool hasAdd = (addend != nullptr);
  #pragma unroll
  for (int ms = 0; ms < 2; ++ms) {
    #pragma unroll
    for (int ns = 0; ns < 2; ++ns) {
      const int c = col0 + wn * 32 + ns * 16 + lrow;
      const float bv = biasb ? biasb[c] : 0.f;
      #pragma unroll
      for (int r = 0; r < 8; ++r) {
        const int m = row0 + wm * 32 + ms * 16 + r + 8 * lgrp;
        float v = acc[ms][ns][r] + bv;
        if (hasAdd) v += addend[(long)m * ldadd + c];
        Cb[(long)m * ldc + c] = v;
      }
    }
  }
}

// ---------------------------------------------------------------------------
// BatchNorm over batch (axis 0) + ReLU; one block per column, B=512 rows.
// Deterministic tree reduction.
// ---------------------------------------------------------------------------
__global__ __launch_bounds__(256) void bn_relu_cols_k(
    const float* __restrict__ X, const float* __restrict__ g,
    const float* __restrict__ e, float* __restrict__ Y, int C)
{
  const int c = blockIdx.x;
  const int tid = threadIdx.x;
  const float v0 = X[(long)tid * C + c];
  const float v1 = X[(long)(tid + 256) * C + c];
  __shared__ float ps[256];
  __shared__ float pq[256];
  ps[tid] = v0 + v1;
  pq[tid] = v0 * v0 + v1 * v1;
  __syncthreads();
  for (int s = 128; s > 0; s >>= 1) {
    if (tid < s) { ps[tid] += ps[tid + s]; pq[tid] += pq[tid + s]; }
    __syncthreads();
  }
  __shared__ float mr[2];
  if (tid == 0) {
    const float m = ps[0] * (1.f / 512.f);
    const float var = pq[0] * (1.f / 512.f) - m * m;
    mr[0] = m;
    mr[1] = rsqrtf(var + 1e-5f);
  }
  __syncthreads();
  const float gg = g[c], ee = e[c], m = mr[0], r = mr[1];
  Y[(long)tid * C + c]         = fmaxf((v0 - m) * r * gg + ee, 0.f);
  Y[(long)(tid + 256) * C + c] = fmaxf((v1 - m) * r * gg + ee, 0.f);
}

// ---------------------------------------------------------------------------
// One tree layer, one block per node l: t = tanh(inp @ Wt[l] + bt[l]);
// o = BN_batch(t)*g+e -> nn_out; a = tanh(o@Wa1[l]+ba1)*Wa2[l]+ba2 -> aux_out.
// ---------------------------------------------------------------------------
__global__ __launch_bounds__(256) void tree_layer_k(
    const float* __restrict__ inp, int inRowStride, int nodeStride, int Fin,
    const float* __restrict__ Wt, const float* __restrict__ bt,
    const float* __restrict__ gt, const float* __restrict__ et,
    const float* __restrict__ Wa1, const float* __restrict__ ba1,
    const float* __restrict__ Wa2, const float* __restrict__ ba2,
    float* __restrict__ nn_out, int L, float* __restrict__ aux_out)
{
  const int l = blockIdx.x;
  const int tid = threadIdx.x;
  __shared__ float Wts[24 * 6];
  __shared__ float prm[27];
  __shared__ float ts[512][6];
  __shared__ float mean_s[6], rstd_s[6];

  if (tid < Fin * 6) Wts[tid] = Wt[(long)l * Fin * 6 + tid];
  if (tid < 6) {
    prm[tid]      = bt[l * 6 + tid];
    prm[6 + tid]  = gt[l * 6 + tid];
    prm[12 + tid] = et[l * 6 + tid];
    prm[18 + tid] = Wa1[l * 6 + tid];
  }
  if (tid == 0) { prm[24] = ba1[l]; prm[25] = Wa2[l]; prm[26] = ba2[l]; }
  __syncthreads();

  #pragma unroll
  for (int rep = 0; rep < 2; ++rep) {
    const int b = tid + rep * 256;
    const float* ip = inp + (long)b * inRowStride + (long)l * nodeStride;
    float accv[6];
    #pragma unroll
    for (int h = 0; h < 6; ++h) accv[h] = prm[h];
    for (int fi = 0; fi < Fin; ++fi) {
      const float xv = ip[fi];
      #pragma unroll
      for (int h = 0; h < 6; ++h) accv[h] += xv * Wts[fi * 6 + h];
    }
    #pragma unroll
    for (int h = 0; h < 6; ++h) ts[b][h] = tanhf(accv[h]);
  }
  __syncthreads();
  if (tid < 6) {  // deterministic serial reduction per channel
    float s = 0.f, q = 0.f;
    for (int b = 0; b < 512; ++b) { const float v = ts[b][tid]; s += v; q += v * v; }
    const float m = s * (1.f / 512.f);
    const float var = q * (1.f / 512.f) - m * m;
    mean_s[tid] = m;
    rstd_s[tid] = rsqrtf(var + 1e-5f);
  }
  __syncthreads();
  #pragma unroll
  for (int rep = 0; rep < 2; ++rep) {
    const int b = tid + rep * 256;
    float s = prm[24];
    #pragma unroll
    for (int h = 0; h < 6; ++h) {
      const float o = (ts[b][h] - mean_s[h]) * rstd_s[h] * prm[6 + h] + prm[12 + h];
      nn_out[(long)b * (L * 6) + l * 6 + h] = o;
      s += o * prm[18 + h];
    }
    aux_out[(long)b * L + l] = tanhf(s) * prm[25] + prm[26];
  }
}

// ---------------------------------------------------------------------------
// Top head: top = BN_batch(tanh(fin @ Wtop + btop)); faux = tanh(top@Wauxt+b)*Wout+bout
// Single block, 512 threads (one per batch row).
// ---------------------------------------------------------------------------
__global__ __launch_bounds__(512) void top_k(
    const float* __restrict__ fin,  // (512, 48)
    const float* __restrict__ Wtop, const float* __restrict__ btop,
    const float* __restrict__ gtop, const float* __restrict__ etop,
    const float* __restrict__ Wauxt, const float* __restrict__ bauxt,
    const float* __restrict__ Wout, const float* __restrict__ bout,
    float* __restrict__ top_out, float* __restrict__ faux_out)
{
  __shared__ float Ws[48 * 6];
  __shared__ float ts[512][6];
  __shared__ float prm[27];
  __shared__ float mean_s[6], rstd_s[6];
  const int tid = threadIdx.x;
  if (tid < 288) Ws[tid] = Wtop[tid];
  if (tid < 6) {
    prm[tid] = btop[tid]; prm[6 + tid] = gtop[tid];
    prm[12 + tid] = etop[tid]; prm[18 + tid] = Wauxt[tid];
  }
  if (tid == 0) { prm[24] = bauxt[0]; prm[25] = Wout[0]; prm[26] = bout[0]; }
  __syncthreads();

  const float* f = fin + (long)tid * 48;
  float accv[6];
  #pragma unroll
  for (int h = 0; h < 6; ++h) accv[h] = prm[h];
  for (int j = 0; j < 48; ++j) {
    const float xv = f[j];
    #pragma unroll
    for (int h = 0; h < 6; ++h) accv[h] += xv * Ws[j * 6 + h];
  }
  #pragma unroll
  for (int h = 0; h < 6; ++h) ts[tid][h] = tanhf(accv[h]);
  __syncthreads();
  if (tid < 6) {
    float s = 0.f, q = 0.f;
    for (int b = 0; b < 512; ++b) { const float v = ts[b][tid]; s += v; q += v * v; }
    const float m = s * (1.f / 512.f);
    const float var = q * (1.f / 512.f) - m * m;
    mean_s[tid] = m;
    rstd_s[tid] = rsqrtf(var + 1e-5f);
  }
  __syncthreads();
  float s = prm[24];
  #pragma unroll
  for (int h = 0; h < 6; ++h) {
    const float o = (ts[tid][h] - mean_s[h]) * rstd_s[h] * prm[6 + h] + prm[12 + h];
    top_out[(long)tid * 6 + h] = o;
    s += o * prm[18 + h];
  }
  faux_out[tid] = tanhf(s) * prm[25] + prm[26];
}

// ---------------------------------------------------------------------------
extern "C" void kernel_launch(void* const* d_in, const int* in_sizes, int n_in,
                              void* d_out, int out_size, void* d_ws, size_t ws_size,
                              hipStream_t stream)
{
  (void)in_sizes; (void)n_in; (void)out_size; (void)ws_size;
  // d_in order follows setup_inputs() dict insertion order:
  const float* x    = (const float*)d_in[0];   // (512, 4096)
  const float* Wp   = (const float*)d_in[1];
  const float* bp   = (const float*)d_in[2];
  const float* Wm[3]  = {(const float*)d_in[3],  (const float*)d_in[7],  (const float*)d_in[11]};
  const float* bm[3]  = {(const float*)d_in[4],  (const float*)d_in[8],  (const float*)d_in[12]};
  const float* gm[3]  = {(const float*)d_in[5],  (const float*)d_in[9],  (const float*)d_in[13]};
  const float* em[3]  = {(const float*)d_in[6],  (const float*)d_in[10], (const float*)d_in[14]};
  const float* Wmo  = (const float*)d_in[15];
  const float* bmo  = (const float*)d_in[16];
  const float* Wg   = (const float*)d_in[17];  // (512, 2048, 16)
  const float* bg   = (const float*)d_in[18];  // (512, 16)
  const float* Wt[4];  const float* btr[4]; const float* gtr[4]; const float* etr[4];
  const float* Wa1[4]; const float* ba1[4]; const float* Wa2[4]; const float* ba2[4];
  for (int i = 0; i < 4; ++i) {
    const int base = 19 + i * 8;
    Wt[i]  = (const float*)d_in[base + 0];
    btr[i] = (const float*)d_in[base + 1];
    gtr[i] = (const float*)d_in[base + 2];
    etr[i] = (const float*)d_in[base + 3];
    Wa1[i] = (const float*)d_in[base + 4];
    ba1[i] = (const float*)d_in[base + 5];
    Wa2[i] = (const float*)d_in[base + 6];
    ba2[i] = (const float*)d_in[base + 7];
  }
  const float* Wtop  = (const float*)d_in[51];
  const float* btop  = (const float*)d_in[52];
  const float* gtop  = (const float*)d_in[53];
  const float* etop  = (const float*)d_in[54];
  const float* Wauxt = (const float*)d_in[55];
  const float* bauxt = (const float*)d_in[56];
  const float* Wout  = (const float*)d_in[57];
  const float* bout  = (const float*)d_in[58];

  // Workspace layout (floats)
  float* ws   = (float*)d_ws;
  float* h0   = ws;              // 512*2048
  float* gfo  = ws + 1048576;    // 512*2048
  float* mpre = ws + 2097152;    // up to 512*256
  float* h1   = ws + 2228224;    // 512*128
  float* h2   = ws + 2293760;    // 512*256
  float* h3   = ws + 2424832;    // 512*128
  float* inp0 = ws + 2490368;    // 512*8192

  // Output layout (reference tuple order, flat)
  float* out  = (float*)d_out;
  float* faux = out;                // 512
  float* top  = out + 512;          // 512*6
  float* aux0 = out + 3584;         // 512*512
  float* aux1 = aux0 + 262144;      // 512*128
  float* aux2 = aux1 + 65536;       // 512*32
  float* aux3 = aux2 + 16384;       // 512*8
  float* nn0  = aux3 + 4096;        // 512*512*6
  float* nn1  = nn0 + 1572864;      // 512*128*6
  float* nn2  = nn1 + 393216;       // 512*32*6
  float* nn3  = nn2 + 98304;        // 512*8*6

  const dim3 blk(256);
  // h0 = gene + pert @ Wp + bp       (A = x[:,2048:], addend = x[:,:2048])
  gemm_wmma_f32_k<<<dim3(4, 32, 1), blk, 0, stream>>>(
      x + 2048, 4096, 0, Wp, 2048, 0, 0, bp, 0, x, 4096, h0, 2048, 0, 2048);
  // MLP layer 0: 2048 -> 128, BN + ReLU
  gemm_wmma_f32_k<<<dim3(4, 2, 1), blk, 0, stream>>>(
      h0, 2048, 0, Wm[0], 128, 0, 0, bm[0], 0, nullptr, 0, mpre, 128, 0, 2048);
  bn_relu_cols_k<<<dim3(128), dim3(256), 0, stream>>>(mpre, gm[0], em[0], h1, 128);
  // MLP layer 1: 128 -> 256
  gemm_wmma_f32_k<<<dim3(4, 4, 1), blk, 0, stream>>>(
      h1, 128, 0, Wm[1], 256, 0, 0, bm[1], 0, nullptr, 0, mpre, 256, 0, 128);
  bn_relu_cols_k<<<dim3(256), dim3(256), 0, stream>>>(mpre, gm[1], em[1], h2, 256);
  // MLP layer 2: 256 -> 128
  gemm_wmma_f32_k<<<dim3(4, 2, 1), blk, 0, stream>>>(
      h2, 256, 0, Wm[2], 128, 0, 0, bm[2], 0, nullptr, 0, mpre, 128, 0, 256);
  bn_relu_cols_k<<<dim3(128), dim3(256), 0, stream>>>(mpre, gm[2], em[2], h3, 128);
  // gfo = h3 @ Wmo + bmo : 128 -> 2048
  gemm_wmma_f32_k<<<dim3(4, 32, 1), blk, 0, stream>>>(
      h3, 128, 0, Wmo, 2048, 0, 0, bmo, 0, nullptr, 0, gfo, 2048, 0, 128);
  // inp0[b, l, o] = gfo @ Wg[l] + bg[l] — node-column mode, 4 nodes (64 cols)
  // per z-block: B col c -> Wg + (z*4 + c/16)*32768 + k*16 + c%16
  gemm_wmma_f32_k<<<dim3(4, 1, 128), blk, 0, stream>>>(
      gfo, 2048, 0, Wg, 16, 32768, 1, bg, 64, nullptr, 0, inp0, 8192, 64, 2048);

  // Tree layers
  tree_layer_k<<<dim3(512), dim3(256), 0, stream>>>(
      inp0, 8192, 16, 16, Wt[0], btr[0], gtr[0], etr[0],
      Wa1[0], ba1[0], Wa2[0], ba2[0], nn0, 512, aux0);
  tree_layer_k<<<dim3(128), dim3(256), 0, stream>>>(
      nn0, 3072, 24, 24, Wt[1], btr[1], gtr[1], etr[1],
      Wa1[1], ba1[1], Wa2[1], ba2[1], nn1, 128, aux1);
  tree_layer_k<<<dim3(32), dim3(256), 0, stream>>>(
      nn1, 768, 24, 24, Wt[2], btr[2], gtr[2], etr[2],
      Wa1[2], ba1[2], Wa2[2], ba2[2], nn2, 32, aux2);
  tree_layer_k<<<dim3(8), dim3(256), 0, stream>>>(
      nn2, 192, 24, 24, Wt[3], btr[3], gtr[3], etr[3],
      Wa1[3], ba1[3], Wa2[3], ba2[3], nn3, 8, aux3);

  // Top head
  top_k<<<dim3(1), dim3(512), 0, stream>>>(
      nn3, Wtop, btop, gtop, etop, Wauxt, bauxt, Wout, bout, top, faux);
}